// DeformationGNN_54666343743957
// MI455X (gfx1250) — compile-verified
//
#include <hip/hip_runtime.h>
#include <hip/hip_bf16.h>

typedef float v2f __attribute__((ext_vector_type(2)));
typedef float v8f __attribute__((ext_vector_type(8)));

#define HID 128

// ---------------- degree / normalization ----------------

__global__ void k_init_deg(float* deg, int n) {
    int i = blockIdx.x * blockDim.x + threadIdx.x;
    if (i < n) deg[i] = 1.0f;  // self-loop contributes 1
}

__global__ void k_accum_deg(const long long* __restrict__ ei, float* deg,
                            int n_edges) {
    int e = blockIdx.x * blockDim.x + threadIdx.x;
    if (e < n_edges) {
        int d = (int)ei[(long long)n_edges + e];  // dst row
        atomicAdd(&deg[d], 1.0f);
    }
}

__global__ void k_finalize_deg(float* deg, int n) {
    int i = blockIdx.x * blockDim.x + threadIdx.x;
    if (i < n) {
        float v = deg[i];
        deg[i] = (v > 0.0f) ? __frsqrt_rn(v) : 0.0f;
    }
}

// ---------------- GEMM1: [N,3] @ [3,128] ----------------

__global__ void k_gemm1(const float* __restrict__ x, const float* __restrict__ W1,
                        float* __restrict__ h, int n) {
    int t = blockIdx.x * blockDim.x + threadIdx.x;
    if (t >= n * HID) return;
    int node = t >> 7;       // /128
    int c = t & (HID - 1);
    float acc = x[node * 3 + 0] * W1[0 * HID + c]
              + x[node * 3 + 1] * W1[1 * HID + c]
              + x[node * 3 + 2] * W1[2 * HID + c];
    h[t] = acc;
}

// ---------------- zero ----------------

__global__ void k_zero(float* p, int n) {
    int i = blockIdx.x * blockDim.x + threadIdx.x;
    if (i < n) p[i] = 0.0f;
}

// ---------------- edge scatter (incl. self-loops) ----------------
// one wave32 per (edge or self-loop); 4 channels per lane

__global__ __launch_bounds__(256)
void k_scatter(const float* __restrict__ hsrc, float* __restrict__ hdst,
               const long long* __restrict__ ei, const float* __restrict__ dis,
               int n_edges, int n_nodes) {
    int e = blockIdx.x * 8 + (threadIdx.x >> 5);
    int lane = threadIdx.x & 31;
    int total = n_edges + n_nodes;
    if (e >= total) return;
    int s, d;
    if (e < n_edges) {
        s = (int)ei[e];
        d = (int)ei[(long long)n_edges + e];
    } else {
        s = d = e - n_edges;  // self-loop
    }
    float norm = dis[s] * dis[d];
    const float4* src4 = (const float4*)(hsrc + (long long)s * HID);
    float4 v = src4[lane];  // channels lane*4 .. lane*4+3
    float* dp = hdst + (long long)d * HID + lane * 4;
    atomicAdd(dp + 0, v.x * norm);
    atomicAdd(dp + 1, v.y * norm);
    atomicAdd(dp + 2, v.z * norm);
    atomicAdd(dp + 3, v.w * norm);
}

// ---------------- bias + ReLU in place ----------------

__global__ void k_bias_relu(float* __restrict__ h, const float* __restrict__ b, int n) {
    int t = blockIdx.x * blockDim.x + threadIdx.x;
    if (t >= n * HID) return;
    float v = h[t] + b[t & (HID - 1)];
    h[t] = v > 0.0f ? v : 0.0f;
}

// ---------------- GEMM2 via WMMA f32 16x16x4 ----------------
// A: [N,128] row-major (relu'd agg1), B: W2 [128,128] row-major, D: [N,128]
// block = 256 threads = 8 waves; block covers 16 rows x 128 cols
// wave w handles the 16x16 tile at columns w*16

__global__ __launch_bounds__(256)
void k_gemm2_wmma(const float* __restrict__ A, const float* __restrict__ B,
                  float* __restrict__ D) {
    int m0 = blockIdx.x * 16;
    int wave = threadIdx.x >> 5;       // 0..7 -> N tile
    int n0 = wave * 16;
    int lane = threadIdx.x & 31;
    int half = lane >> 4;              // 0: K pair 0-1, 1: K pair 2-3 (A/B f32 layout)
    int l = lane & 15;                 // A: row index; B/C/D: column index

    const float* Arow = A + (long long)(m0 + l) * HID;

    v8f acc = {};
#pragma unroll
    for (int kk = 0; kk < HID; kk += 4) {
        int ka = kk + half * 2;
        v2f a, b;
        a.x = Arow[ka];
        a.y = Arow[ka + 1];
        b.x = B[ka * HID + n0 + l];
        b.y = B[(ka + 1) * HID + n0 + l];
        acc = __builtin_amdgcn_wmma_f32_16x16x4_f32(
            /*neg_a=*/false, a, /*neg_b=*/false, b,
            /*c_mod=*/(short)0, acc, /*reuse_a=*/false, /*reuse_b=*/false);
    }
    // D layout: VGPR r holds row m0 + r + half*8, column n0 + l
#pragma unroll
    for (int r = 0; r < 8; ++r) {
        D[(long long)(m0 + r + half * 8) * HID + n0 + l] = acc[r];
    }
}

// ---------------- FC: [N,128] @ [128,3] + bias ----------------

__global__ void k_fc(const float* __restrict__ h, const float* __restrict__ Wfc,
                     const float* __restrict__ bfc, float* __restrict__ out, int n) {
    int t = blockIdx.x * blockDim.x + threadIdx.x;
    if (t >= n * 3) return;
    int node = t / 3;
    int j = t - node * 3;
    const float* hp = h + (long long)node * HID;
    float acc = bfc[j];
#pragma unroll 4
    for (int c = 0; c < HID; ++c) acc += hp[c] * Wfc[c * 3 + j];
    out[t] = acc;
}

extern "C" void kernel_launch(void* const* d_in, const int* in_sizes, int n_in,
                              void* d_out, int out_size, void* d_ws, size_t ws_size,
                              hipStream_t stream) {
    const float*      x   = (const float*)d_in[0];
    const long long*  ei  = (const long long*)d_in[1];   // int64 [2, E]
    const float*      W1  = (const float*)d_in[2];
    const float*      b1  = (const float*)d_in[3];
    const float*      W2  = (const float*)d_in[4];
    const float*      b2  = (const float*)d_in[5];
    const float*      Wfc = (const float*)d_in[6];
    const float*      bfc = (const float*)d_in[7];
    float* out = (float*)d_out;

    const int N = in_sizes[0] / 3;       // 100000
    const int E = in_sizes[1] / 2;       // 3200000

    // workspace: dis [N], hA [N*128], hB [N*128]
    char* ws = (char*)d_ws;
    size_t off_dis = 0;
    size_t off_hA  = ((size_t)N * 4 + 511) & ~(size_t)511;
    size_t off_hB  = off_hA + (size_t)N * HID * 4;
    float* dis = (float*)(ws + off_dis);
    float* hA  = (float*)(ws + off_hA);
    float* hB  = (float*)(ws + off_hB);

    const int T = 256;
    const int nh = N * HID;

    // ---- degree / dis ----
    k_init_deg<<<(N + T - 1) / T, T, 0, stream>>>(dis, N);
    k_accum_deg<<<(E + T - 1) / T, T, 0, stream>>>(ei, dis, E);
    k_finalize_deg<<<(N + T - 1) / T, T, 0, stream>>>(dis, N);

    // ---- layer 1: h = x@W1 ; aggregate ; +b1 ; relu ----
    k_gemm1<<<(nh + T - 1) / T, T, 0, stream>>>(x, W1, hA, N);
    k_zero<<<(nh + T - 1) / T, T, 0, stream>>>(hB, nh);
    {
        int total = E + N;
        k_scatter<<<(total + 7) / 8, T, 0, stream>>>(hA, hB, ei, dis, E, N);
    }
    k_bias_relu<<<(nh + T - 1) / T, T, 0, stream>>>(hB, b1, N);

    // ---- layer 2: h = relu1@W2 (WMMA) ; aggregate ; +b2 ; relu ----
    k_gemm2_wmma<<<N / 16, T, 0, stream>>>(hB, W2, hA);
    k_zero<<<(nh + T - 1) / T, T, 0, stream>>>(hB, nh);
    {
        int total = E + N;
        k_scatter<<<(total + 7) / 8, T, 0, stream>>>(hA, hB, ei, dis, E, N);
    }
    k_bias_relu<<<(nh + T - 1) / T, T, 0, stream>>>(hB, b2, N);

    // ---- FC head ----
    k_fc<<<(N * 3 + T - 1) / T, T, 0, stream>>>(hB, Wfc, bfc, out, N);
}